// SelfAttentionLayer_43525198577907
// MI455X (gfx1250) — compile-verified
//
#include <hip/hip_runtime.h>
#include <math.h>

#define B_   4
#define S_   4096
#define H_   2048
#define HD2  1024
#define SCALE_   0.022097086912079608f   // 1/sqrt(2048)
#define LN_BASE_ 6.907755278982137f      // ln(1000)
#define NEG_INF_ -1e30f
#define APAD 2052   // padded LDS row stride (floats): b64 lane banks (4r+k)%64 all distinct
#define MROWS 32    // rows of hs per workgroup in the scores kernel

typedef __attribute__((ext_vector_type(2))) float v2f;
typedef __attribute__((ext_vector_type(8))) float v8f;
typedef __attribute__((__vector_size__(4 * sizeof(int)))) int v4i_;

// ---------- Kernel 1: Q projection (GEMV: one wave per output channel) ----------
__global__ void qproj_kernel(const float* __restrict__ hs, const float* __restrict__ Wq,
                             const float* __restrict__ bq, const int* __restrict__ idx,
                             float* __restrict__ qraw) {
  int gtid = blockIdx.x * blockDim.x + threadIdx.x;
  int wave = gtid >> 5;
  int lane = threadIdx.x & 31;
  int b = wave >> 11;            // 2048 waves per batch
  int n = wave & (H_ - 1);
  int row = idx[b] - 1;
  const float* x = hs + ((size_t)b * S_ + row) * H_;
  const float* w = Wq + (size_t)n * H_;
  float acc = 0.f;
  for (int h = lane * 4; h < H_; h += 128) {
    float4 xv = *(const float4*)(x + h);
    float4 wv = *(const float4*)(w + h);
    acc += xv.x * wv.x + xv.y * wv.y + xv.z * wv.z + xv.w * wv.w;
  }
  for (int off = 16; off > 0; off >>= 1) acc += __shfl_down(acc, off);
  if (lane == 0) qraw[b * H_ + n] = acc + bq[n];
}

// ---------- Kernel 1b: RoPE on the selected query + pre-scale by 1/sqrt(H) ----------
__global__ void qrope_kernel(const float* __restrict__ qraw, const int* __restrict__ idx,
                             float* __restrict__ qrot) {
  int t = blockIdx.x * blockDim.x + threadIdx.x;   // B*H threads
  int b = t >> 11;
  int n = t & (H_ - 1);
  float pos = (float)(idx[b] - 1);
  int j = n & (HD2 - 1);
  float f = expf(-(float)(2 * j) * (LN_BASE_ / (float)H_));
  float sn, cs;
  sincosf(pos * f, &sn, &cs);
  float qn = qraw[t];
  float rot = (n < HD2) ? -qraw[t + HD2] : qraw[t - HD2];
  qrot[t] = (qn * cs + rot * sn) * SCALE_;
}

// Fold K-side RoPE into a per-(s,n) query weight, add bk contribution, accumulate.
__device__ __forceinline__ void accum_scores(const v8f& cT0, const v8f& cT1, int n, int b,
                                             int s0, int lhalf,
                                             const float* __restrict__ qrot,
                                             const float* __restrict__ bk,
                                             float* part0, float* part1) {
  float qn  = qrot[b * H_ + n];
  float qp  = (n < HD2) ? qrot[b * H_ + n + HD2] : qrot[b * H_ + n - HD2];
  float sgn = (n < HD2) ? 1.f : -1.f;
  int   j   = n & (HD2 - 1);
  float f   = expf(-(float)(2 * j) * (LN_BASE_ / (float)H_));
  float bkn = bk[n];
#pragma unroll
  for (int v = 0; v < 8; ++v) {
    int m = v + 8 * lhalf;                 // C layout: lanes 16-31 hold M=8..15
    float sn, cs;
    sincosf((float)(s0 + m) * f, &sn, &cs);
    part0[v] += (cT0[v] + bkn) * (qn * cs + sgn * qp * sn);
    sincosf((float)(s0 + 16 + m) * f, &sn, &cs);
    part1[v] += (cT1[v] + bkn) * (qn * cs + sgn * qp * sn);
  }
}

// ---------- Kernel 2: scores[b,s] = (hs_tile @ Wk^T + bk) · w_rope(s,·)  (WMMA f32) ----------
// Grid: (S/32, B). Block: 256 threads (8 waves). Each wave: 2x2 register-blocked WMMA
// (2 m-tiles x 2 n-tiles share operand fetches -> halves LDS and L2 operand traffic).
// LDS: 32x2048 hs tile (async-staged, padded) + 8x32 partials.
__global__ void __launch_bounds__(256)
scores_kernel(const float* __restrict__ hs, const float* __restrict__ Wk,
              const float* __restrict__ bk, const float* __restrict__ qrot,
              float* __restrict__ scores) {
  extern __shared__ float smem[];
  float* sA = smem;                       // MROWS * APAD floats
  float* sPart = smem + MROWS * APAD;     // 8 waves * 32 m-values
  const int b  = blockIdx.y;
  const int s0 = blockIdx.x * MROWS;
  const int tid = threadIdx.x;

  // Stage A tile: 32 rows of hs, 2048 cols each, as 16B chunks.
  {
    const float* srcBase = hs + ((size_t)b * S_ + s0) * H_;
#if __has_builtin(__builtin_amdgcn_global_load_async_to_lds_b128)
    for (int it = 0; it < (MROWS * (H_ / 4)) / 256; ++it) {   // 64 iterations
      int q  = it * 256 + tid;
      int r  = q >> 9;          // 512 float4-chunks per row
      int c4 = q & 511;
      const float* gp = srcBase + (size_t)r * H_ + c4 * 4;
      float* lp = sA + r * APAD + c4 * 4;
      __builtin_amdgcn_global_load_async_to_lds_b128((v4i_*)gp, (v4i_*)lp, 0, 0);
    }
#if __has_builtin(__builtin_amdgcn_s_wait_asynccnt)
    __builtin_amdgcn_s_wait_asynccnt(0);
#else
    asm volatile("s_wait_asynccnt 0" ::: "memory");
#endif
#else
    for (int it = 0; it < (MROWS * (H_ / 4)) / 256; ++it) {
      int q  = it * 256 + tid;
      int r  = q >> 9;
      int c4 = q & 511;
      *(float4*)(sA + r * APAD + c4 * 4) = *(const float4*)(srcBase + (size_t)r * H_ + c4 * 4);
    }
#endif
  }
  __syncthreads();

  const int wave  = tid >> 5;
  const int lane  = tid & 31;
  const int lmod  = lane & 15;   // = N within tile (for B/C), = M row (for A)
  const int lhalf = lane >> 4;   // A/B: K-half selector; C: M-half selector
  const float* aBase0 = sA + lmod * APAD + 2 * lhalf;            // m-tile 0 (rows 0..15)
  const float* aBase1 = aBase0 + 16 * APAD;                      // m-tile 1 (rows 16..31)

  float part0[8], part1[8];
#pragma unroll
  for (int v = 0; v < 8; ++v) { part0[v] = 0.f; part1[v] = 0.f; }

  // 16 n-tiles per wave, processed as 8 adjacent pairs.
  for (int p = 0; p < 8; ++p) {
    const int nA = (wave * 16 + p * 2) * 16 + lmod;
    const int nB = nA + 16;
    const float* bA = Wk + (size_t)nA * H_ + 2 * lhalf;
    const float* bB = bA + (size_t)16 * H_;
    v8f c00 = {0.f,0.f,0.f,0.f,0.f,0.f,0.f,0.f};
    v8f c01 = c00, c10 = c00, c11 = c00;
#pragma unroll 2
    for (int k0 = 0; k0 < H_; k0 += 4) {
      v2f a0 = *(const v2f*)(aBase0 + k0);
      v2f a1 = *(const v2f*)(aBase1 + k0);
      v2f b0 = *(const v2f*)(bA + k0);
      v2f b1 = *(const v2f*)(bB + k0);
      c00 = __builtin_amdgcn_wmma_f32_16x16x4_f32(false, a0, false, b0, (short)0, c00, false, false);
      c01 = __builtin_amdgcn_wmma_f32_16x16x4_f32(false, a0, false, b1, (short)0, c01, false, false);
      c10 = __builtin_amdgcn_wmma_f32_16x16x4_f32(false, a1, false, b0, (short)0, c10, false, false);
      c11 = __builtin_amdgcn_wmma_f32_16x16x4_f32(false, a1, false, b1, (short)0, c11, false, false);
    }
    accum_scores(c00, c10, nA, b, s0, lhalf, qrot, bk, part0, part1);
    accum_scores(c01, c11, nB, b, s0, lhalf, qrot, bk, part0, part1);
  }

  // Reduce over N: sum across the 16 lanes of each half-wave.
#pragma unroll
  for (int v = 0; v < 8; ++v) {
    float p0 = part0[v], p1 = part1[v];
    p0 += __shfl_xor(p0, 1); p1 += __shfl_xor(p1, 1);
    p0 += __shfl_xor(p0, 2); p1 += __shfl_xor(p1, 2);
    p0 += __shfl_xor(p0, 4); p1 += __shfl_xor(p1, 4);
    p0 += __shfl_xor(p0, 8); p1 += __shfl_xor(p1, 8);
    part0[v] = p0; part1[v] = p1;
  }
  if (lmod == 0) {
#pragma unroll
    for (int v = 0; v < 8; ++v) {
      sPart[wave * 32 + 8 * lhalf + v]      = part0[v];   // rows 0..15
      sPart[wave * 32 + 16 + 8 * lhalf + v] = part1[v];   // rows 16..31
    }
  }
  __syncthreads();
  if (tid < MROWS) {
    float s = 0.f;
    for (int w = 0; w < 8; ++w) s += sPart[w * 32 + tid];
    scores[b * S_ + s0 + tid] = s;
  }
}

// ---------- Kernel 3: masked softmax per batch ----------
__global__ void softmax_kernel(const float* __restrict__ scores, const int* __restrict__ amask,
                               const int* __restrict__ idx, float* __restrict__ attn) {
  const int b = blockIdx.x;
  const int tid = threadIdx.x;
  const int limit = idx[b];
  __shared__ float red[8];
  float vals[16];
  float lmax = -3.4e38f;
#pragma unroll
  for (int i = 0; i < 16; ++i) {
    int s = tid + i * 256;
    float sc = scores[b * S_ + s];
    bool kept = (amask[b * S_ + s] != 0) && (s < limit);
    float lg = kept ? sc : NEG_INF_;
    vals[i] = lg;
    lmax = fmaxf(lmax, lg);
  }
  for (int off = 16; off > 0; off >>= 1) lmax = fmaxf(lmax, __shfl_xor(lmax, off));
  if ((tid & 31) == 0) red[tid >> 5] = lmax;
  __syncthreads();
  float gmax = red[0];
  for (int w = 1; w < 8; ++w) gmax = fmaxf(gmax, red[w]);
  __syncthreads();
  float lsum = 0.f;
#pragma unroll
  for (int i = 0; i < 16; ++i) { vals[i] = expf(vals[i] - gmax); lsum += vals[i]; }
  for (int off = 16; off > 0; off >>= 1) lsum += __shfl_xor(lsum, off);
  if ((tid & 31) == 0) red[tid >> 5] = lsum;
  __syncthreads();
  float gsum = 0.f;
  for (int w = 0; w < 8; ++w) gsum += red[w];
  float inv = 1.f / gsum;
#pragma unroll
  for (int i = 0; i < 16; ++i) attn[b * S_ + tid + i * 256] = vals[i] * inv;
}

// ---------- Kernel 4: zero + r = attn @ hs (memory-bound pass over hs) ----------
__global__ void zero_kernel(float* __restrict__ p, int n) {
  int t = blockIdx.x * blockDim.x + threadIdx.x;
  if (t < n) p[t] = 0.f;
}

__global__ void attnv_kernel(const float* __restrict__ hs, const float* __restrict__ attn,
                             float* __restrict__ r) {
  const int b = blockIdx.z;
  const int h = blockIdx.x * 256 + threadIdx.x;
  const int sbase = blockIdx.y * 128;
  const float* hp = hs + ((size_t)b * S_ + sbase) * H_ + h;
  const float* ap = attn + b * S_ + sbase;
  float acc = 0.f;
  for (int i = 0; i < 128; ++i) acc += ap[i] * hp[(size_t)i * H_];
  atomicAdd(&r[b * H_ + h], acc);
}

// ---------- Kernel 5: out = r @ Wv^T + bv  (V-GEMM eliminated by associativity) ----------
__global__ void outproj_kernel(const float* __restrict__ r, const float* __restrict__ Wv,
                               const float* __restrict__ bv, float* __restrict__ out) {
  int gtid = blockIdx.x * blockDim.x + threadIdx.x;
  int wave = gtid >> 5;
  int lane = threadIdx.x & 31;
  int b = wave >> 11;
  int n = wave & (H_ - 1);
  const float* x = r + b * H_;
  const float* w = Wv + (size_t)n * H_;
  float acc = 0.f;
  for (int h = lane * 4; h < H_; h += 128) {
    float4 xv = *(const float4*)(x + h);
    float4 wv = *(const float4*)(w + h);
    acc += xv.x * wv.x + xv.y * wv.y + xv.z * wv.z + xv.w * wv.w;
  }
  for (int off = 16; off > 0; off >>= 1) acc += __shfl_down(acc, off);
  if (lane == 0) out[b * H_ + n] = acc + bv[n];
}

extern "C" void kernel_launch(void* const* d_in, const int* in_sizes, int n_in,
                              void* d_out, int out_size, void* d_ws, size_t ws_size,
                              hipStream_t stream) {
  const float* hs = (const float*)d_in[0];
  const float* Wq = (const float*)d_in[1];
  const float* bq = (const float*)d_in[2];
  const float* Wk = (const float*)d_in[3];
  const float* bk = (const float*)d_in[4];
  const float* Wv = (const float*)d_in[5];
  const float* bv = (const float*)d_in[6];
  const int* amask = (const int*)d_in[7];
  const int* idx   = (const int*)d_in[8];
  float* out = (float*)d_out;

  float* ws = (float*)d_ws;
  float* qraw   = ws;                 // B*H
  float* qrot   = ws + 8192;          // B*H
  float* scores = ws + 16384;         // B*S
  float* attn   = ws + 32768;         // B*S
  float* r      = ws + 49152;         // B*H

  // 1) Q projection for the 4 selected rows, then RoPE+scale.
  qproj_kernel<<<1024, 256, 0, stream>>>(hs, Wq, bq, idx, qraw);
  qrope_kernel<<<32, 256, 0, stream>>>(qraw, idx, qrot);

  // 2) Fused K-projection + RoPE-folded score GEMM (WMMA f32 16x16x4, 2x2 blocked).
  size_t lds_bytes = (size_t)(MROWS * APAD + 8 * 32) * sizeof(float);
  scores_kernel<<<dim3(S_ / MROWS, B_), 256, lds_bytes, stream>>>(hs, Wk, bk, qrot, scores);

  // 3) Masked softmax per batch.
  softmax_kernel<<<B_, 256, 0, stream>>>(scores, amask, idx, attn);

  // 4) r = attn @ hs  (V projection deferred past the reduction).
  zero_kernel<<<(B_ * H_ + 255) / 256, 256, 0, stream>>>(r, B_ * H_);
  attnv_kernel<<<dim3(H_ / 256, S_ / 128, B_), 256, 0, stream>>>(hs, attn, r);

  // 5) out = r @ Wv^T + bv.
  outproj_kernel<<<1024, 256, 0, stream>>>(r, Wv, bv, out);
}